// QLSTM_65481071399249
// MI455X (gfx1250) — compile-verified
//
#include <hip/hip_runtime.h>
#include <math.h>

typedef __attribute__((ext_vector_type(2))) float v2f;
typedef __attribute__((ext_vector_type(4))) float v4f;
typedef __attribute__((ext_vector_type(8))) float v8f;

#define NQ      10
#define NSTATE  1024
#define BATCH   256
#define DIN     32
#define TSTEPS  64
#define SROWS   1024   /* 4 gates * BATCH rows per step */

// ---------------------------------------------------------------------------
// zero hx / cx carry (they live in the tail of d_out)
// ---------------------------------------------------------------------------
__global__ void k_init(float* hc) {
  int i = blockIdx.x * blockDim.x + threadIdx.x;
  if (i < 2 * BATCH * NQ) hc[i] = 0.f;
}

// ---------------------------------------------------------------------------
// Precompute the fixed post-encoding circuit unitary U (1024x1024 complex),
// stored transposed: UT[k][j] = U[j][k].  One workgroup per column k,
// statevector in LDS.  Qubit w <-> flat-index bit (9-w)  (axis order of the
// reference's (2,)*n state shape).
// ---------------------------------------------------------------------------
__global__ __launch_bounds__(128)
void k_buildU(const float* __restrict__ qp,
              float* __restrict__ UTre, float* __restrict__ UTim) {
  __shared__ float sr[NSTATE];
  __shared__ float si[NSTATE];
  int tid = threadIdx.x;
  int col = blockIdx.x;
  for (int i = tid; i < NSTATE; i += 128) { sr[i] = 0.f; si[i] = 0.f; }
  __syncthreads();
  if (tid == 0) sr[col] = 1.f;

  for (int l = 0; l < 2; ++l) {
    for (int w = 0; w < NQ; ++w) {
      float t0 = qp[(l * NQ + w) * 3 + 0];
      float t1 = qp[(l * NQ + w) * 3 + 1];
      float t2 = qp[(l * NQ + w) * 3 + 2];
      float cz = cosf(0.5f * t0), sz = sinf(0.5f * t0);
      float c1 = cosf(0.5f * t1), s1 = sinf(0.5f * t1);
      float c2 = cosf(0.5f * t2), s2 = sinf(0.5f * t2);
      // A = RX(t1) * RZ(t0)
      float A00r =  c1 * cz, A00i = -c1 * sz;
      float A01r =  s1 * sz, A01i = -s1 * cz;
      float A10r = -s1 * sz, A10i = -s1 * cz;
      float A11r =  c1 * cz, A11i =  c1 * sz;
      // M = RY(t2) * A   (reference applies RZ, then RX, then RY)
      float M00r = c2 * A00r - s2 * A10r, M00i = c2 * A00i - s2 * A10i;
      float M01r = c2 * A01r - s2 * A11r, M01i = c2 * A01i - s2 * A11i;
      float M10r = s2 * A00r + c2 * A10r, M10i = s2 * A00i + c2 * A10i;
      float M11r = s2 * A01r + c2 * A11r, M11i = s2 * A01i + c2 * A11i;
      int b = 9 - w;
      __syncthreads();
      for (int pp = 0; pp < 4; ++pp) {
        int p  = tid + 128 * pp;
        int i0 = ((p >> b) << (b + 1)) | (p & ((1 << b) - 1));
        int i1 = i0 | (1 << b);
        float ar = sr[i0], ai = si[i0], br = sr[i1], bi = si[i1];
        sr[i0] = M00r * ar - M00i * ai + M01r * br - M01i * bi;
        si[i0] = M00r * ai + M00i * ar + M01r * bi + M01i * br;
        sr[i1] = M10r * ar - M10i * ai + M11r * br - M11i * bi;
        si[i1] = M10r * ai + M10i * ar + M11r * bi + M11i * br;
      }
    }
    // CNOT ring: (0,1)(1,2)...(8,9)(9,0)
    for (int e = 0; e < NQ; ++e) {
      int c  = (e < 9) ? e : 9;
      int t  = (e < 9) ? e + 1 : 0;
      int bc = 9 - c, bt = 9 - t;
      int blo = bc < bt ? bc : bt;
      int bhi = bc < bt ? bt : bc;
      __syncthreads();
      for (int pp = 0; pp < 2; ++pp) {
        int p = tid + 128 * pp;
        int q = ((p >> blo) << (blo + 1)) | (p & ((1 << blo) - 1));
        q     = ((q >> bhi) << (bhi + 1)) | (q & ((1 << bhi) - 1));
        int iA = q | (1 << bc);        // control bit = 1, target bit = 0
        int iB = iA | (1 << bt);
        float tr = sr[iA], ti = si[iA];
        sr[iA] = sr[iB]; si[iA] = si[iB];
        sr[iB] = tr;     si[iB] = ti;
      }
    }
  }
  __syncthreads();
  for (int j = tid; j < NSTATE; j += 128) {
    UTre[col * NSTATE + j] = sr[j];
    UTim[col * NSTATE + j] = si[j];
  }
}

// ---------------------------------------------------------------------------
// Per timestep: gate pre-activations + real product-state encoding.
// One block per circuit row s (1024).  Also zeroes exps[s][*].
// ---------------------------------------------------------------------------
__global__ __launch_bounds__(128)
void k_enc(int t, const float* __restrict__ xin, const float* __restrict__ hx,
           const float* __restrict__ Wf, const float* __restrict__ bf,
           const float* __restrict__ Wi, const float* __restrict__ bi,
           const float* __restrict__ Wg, const float* __restrict__ bg,
           const float* __restrict__ Wo, const float* __restrict__ bo,
           float* __restrict__ enc, float* __restrict__ exps) {
  __shared__ float cb[NQ], sb[NQ];
  __shared__ float lo[32], hi[32];
  int tid  = threadIdx.x;
  int s    = blockIdx.x;
  int gate = s >> 8;
  int b    = s & 255;
  if (tid < NQ) {
    const float* W; const float* bs;
    if      (gate == 0) { W = Wf; bs = bf; }
    else if (gate == 1) { W = Wi; bs = bi; }
    else if (gate == 2) { W = Wg; bs = bg; }
    else                { W = Wo; bs = bo; }
    const float* wr = W + tid * 42;
    float acc = bs[tid];
    const float* xr = xin + (t * BATCH + b) * DIN;
    for (int j = 0; j < DIN; ++j) acc += xr[j] * wr[j];
    const float* hr = hx + b * NQ;
    for (int j = 0; j < NQ; ++j) acc += hr[j] * wr[DIN + j];
    float h   = 0.5f * acc;
    int   bit = 9 - tid;               // qubit tid -> flat bit 9-tid
    cb[bit] = cosf(h);
    sb[bit] = sinf(h);
    exps[s * NQ + tid] = 0.f;
  }
  __syncthreads();
  if (tid < 32) {
    float v = 1.f;
    for (int bit = 0; bit < 5; ++bit) v *= ((tid >> bit) & 1) ? sb[bit] : cb[bit];
    lo[tid] = v;
  } else if (tid < 64) {
    int u = tid - 32;
    float v = 1.f;
    for (int bit = 0; bit < 5; ++bit) v *= ((u >> bit) & 1) ? sb[5 + bit] : cb[5 + bit];
    hi[u] = v;
  }
  __syncthreads();
  float* er = enc + s * NSTATE;
  for (int i = 0; i < 8; ++i) {
    int k = tid * 8 + i;
    er[k] = lo[k & 31] * hi[k >> 5];
  }
}

// ---------------------------------------------------------------------------
// Main WMMA GEMM + fused Z-measurement.
// out = enc(1024xK) @ UT(Kx1024), complex via two f32 WMMA accumulator chains.
// grid (64,16), 128 threads (4 waves); wave owns one 16x16 (re,im) tile pair.
// Epilogue: |amp|^2, 16-wide Walsh-Hadamard over low 4 bits of j, LDS + global
// atomic accumulation into exps[1024][10].
// ---------------------------------------------------------------------------
__global__ __launch_bounds__(128)
void k_gemm(const float* __restrict__ enc, const float* __restrict__ UTre,
            const float* __restrict__ UTim, float* __restrict__ exps) {
  __shared__ float sA[16][68];
  __shared__ float sBr[64][68];   // [n][k] transposed
  __shared__ float sBi[64][68];
  __shared__ float accS[16][NQ];
  int tid  = threadIdx.x;
  int m0   = blockIdx.x * 16;
  int nblk = blockIdx.y * 64;
  int lane = tid & 31;
  int wid  = tid >> 5;
  int half = lane >> 4;
  int row  = lane & 15;
  int ncol = wid * 16 + row;

  if (tid < 16 * NQ) accS[tid / NQ][tid % NQ] = 0.f;

  v8f accR = {0.f, 0.f, 0.f, 0.f, 0.f, 0.f, 0.f, 0.f};
  v8f accI = {0.f, 0.f, 0.f, 0.f, 0.f, 0.f, 0.f, 0.f};

  for (int kc = 0; kc < NSTATE; kc += 64) {
    __syncthreads();
    { // stage A chunk: 16 x 64
      int f = tid * 8;
      int r = f >> 6, c = f & 63;
      const v4f* g0 = (const v4f*)(enc + (m0 + r) * NSTATE + kc + c);
      v4f a0 = g0[0], a1 = g0[1];
      sA[r][c + 0] = a0.x; sA[r][c + 1] = a0.y; sA[r][c + 2] = a0.z; sA[r][c + 3] = a0.w;
      sA[r][c + 4] = a1.x; sA[r][c + 5] = a1.y; sA[r][c + 6] = a1.z; sA[r][c + 7] = a1.w;
    }
    for (int i = 0; i < 8; ++i) { // stage B chunk: 64 x 64 (re & im), transpose
      int f = (tid * 8 + i) * 4;
      int k = f >> 6, n = f & 63;
      v4f br = *(const v4f*)(UTre + (kc + k) * NSTATE + nblk + n);
      v4f bi = *(const v4f*)(UTim + (kc + k) * NSTATE + nblk + n);
      sBr[n + 0][k] = br.x; sBr[n + 1][k] = br.y; sBr[n + 2][k] = br.z; sBr[n + 3][k] = br.w;
      sBi[n + 0][k] = bi.x; sBi[n + 1][k] = bi.y; sBi[n + 2][k] = bi.z; sBi[n + 3][k] = bi.w;
    }
    __syncthreads();
    for (int ks = 0; ks < 64; ks += 4) {
      // A frag: lanes 0-15 hold rows, K={ks,ks+1}; lanes 16-31 K={ks+2,ks+3}
      v2f a  = *(const v2f*)&sA [row ][ks + 2 * half];
      v2f br = *(const v2f*)&sBr[ncol][ks + 2 * half];
      v2f bi = *(const v2f*)&sBi[ncol][ks + 2 * half];
      accR = __builtin_amdgcn_wmma_f32_16x16x4_f32(false, a, false, br, (short)0, accR, false, false);
      accI = __builtin_amdgcn_wmma_f32_16x16x4_f32(false, a, false, bi, (short)0, accI, false, false);
    }
  }

  // fused measurement: exps[s][w] += sum_j sign_w(j) * |out[s][j]|^2
  int n0full = nblk + wid * 16;        // upper bits of j for this tile
  int tlo    = lane & 15;
  for (int r = 0; r < 8; ++r) {
    float p = accR[r] * accR[r] + accI[r] * accI[r];
    float v = p;
    for (int k = 0; k < 4; ++k) {      // 16-wide Walsh-Hadamard butterfly
      float o = __shfl_xor(v, 1 << k, 32);
      v = ((lane >> k) & 1) ? (o - v) : (o + v);
    }
    int m = r + 8 * half;              // row inside tile
    if      (tlo == 1) atomicAdd(&accS[m][9], v);   // j bit0 <-> qubit 9
    else if (tlo == 2) atomicAdd(&accS[m][8], v);
    else if (tlo == 4) atomicAdd(&accS[m][7], v);
    else if (tlo == 8) atomicAdd(&accS[m][6], v);
    else if (tlo == 0) {
      for (int w = 0; w < 6; ++w) {    // bits 4..9 constant inside tile
        float sg = ((n0full >> (9 - w)) & 1) ? -v : v;
        atomicAdd(&accS[m][w], sg);
      }
    }
  }
  __syncthreads();
  if (tid < 16 * NQ) {
    int m = tid / NQ, w = tid % NQ;
    atomicAdd(&exps[(m0 + m) * NQ + w], accS[m][w]);
  }
}

// ---------------------------------------------------------------------------
// LSTM cell update + output write.
// ---------------------------------------------------------------------------
__global__ void k_final(int t, const float* __restrict__ exps,
                        float* __restrict__ outp,
                        float* __restrict__ hx, float* __restrict__ cx) {
  int i = blockIdx.x * blockDim.x + threadIdx.x;
  if (i >= BATCH * NQ) return;
  int b = i / NQ, w = i % NQ;
  float qf = exps[(0 * BATCH + b) * NQ + w];
  float qi = exps[(1 * BATCH + b) * NQ + w];
  float qg = exps[(2 * BATCH + b) * NQ + w];
  float qo = exps[(3 * BATCH + b) * NQ + w];
  float f  = 1.f / (1.f + expf(-qf));
  float ii = 1.f / (1.f + expf(-qi));
  float g  = tanhf(qg);
  float o  = 1.f / (1.f + expf(-qo));
  float c2 = f * cx[i] + ii * g;
  float h2 = o * tanhf(c2);
  cx[i] = c2;
  hx[i] = h2;
  outp[t * BATCH * NQ + i] = h2;
}

// ---------------------------------------------------------------------------
extern "C" void kernel_launch(void* const* d_in, const int* in_sizes, int n_in,
                              void* d_out, int out_size, void* d_ws, size_t ws_size,
                              hipStream_t stream) {
  (void)in_sizes; (void)n_in; (void)out_size; (void)ws_size;
  const float* xin = (const float*)d_in[0];
  const float* qp  = (const float*)d_in[1];
  const float* Wf  = (const float*)d_in[2];
  const float* bf  = (const float*)d_in[3];
  const float* Wi  = (const float*)d_in[4];
  const float* bi  = (const float*)d_in[5];
  const float* Wg  = (const float*)d_in[6];
  const float* bg  = (const float*)d_in[7];
  const float* Wo  = (const float*)d_in[8];
  const float* bo  = (const float*)d_in[9];

  float* out = (float*)d_out;
  float* hx  = out + TSTEPS * BATCH * NQ;   // carries live in d_out tail
  float* cx  = hx + BATCH * NQ;

  float* ws   = (float*)d_ws;
  float* UTre = ws;                           // 1024*1024
  float* UTim = UTre + NSTATE * NSTATE;       // 1024*1024
  float* enc  = UTim + NSTATE * NSTATE;       // 1024*1024
  float* exps = enc  + SROWS * NSTATE;        // 1024*10

  k_init  <<<dim3((2 * BATCH * NQ + 255) / 256), 256, 0, stream>>>(hx);
  k_buildU<<<dim3(NSTATE), 128, 0, stream>>>(qp, UTre, UTim);
  for (int t = 0; t < TSTEPS; ++t) {
    k_enc  <<<dim3(SROWS),   128, 0, stream>>>(t, xin, hx, Wf, bf, Wi, bi, Wg, bg, Wo, bo, enc, exps);
    k_gemm <<<dim3(64, 16),  128, 0, stream>>>(enc, UTre, UTim, exps);
    k_final<<<dim3((BATCH * NQ + 255) / 256), 256, 0, stream>>>(t, exps, out, hx, cx);
  }
}